// LinAttention_KQ_8014408974651
// MI455X (gfx1250) — compile-verified
//
#include <hip/hip_runtime.h>

typedef __attribute__((ext_vector_type(2))) float v2f;
typedef __attribute__((ext_vector_type(4))) float v4f;
typedef __attribute__((ext_vector_type(8))) float v8f;

constexpr int BB = 32, SS = 4096, DD = 128, HH = 8;
constexpr int SCH    = 8;          // S-split for the Gram phase
constexpr int SCHLEN = SS / SCH;   // 512
constexpr int NDD    = DD * DD;    // 16384 elements per DxD matrix
constexpr int NBD    = BB * NDD;   // 524288 elements per [B,D,D] tensor

__device__ __forceinline__ v8f wmma4(v2f a, v2f b, v8f c) {
  // D(16x16 f32) = A(16x4 f32) x B(4x16 f32) + C
  return __builtin_amdgcn_wmma_f32_16x16x4_f32(
      /*neg_a=*/false, a, /*neg_b=*/false, b,
      /*c_mod=*/(short)0, c, /*reuse_a=*/false, /*reuse_b=*/false);
}

// ---------------------------------------------------------------------------
// Phase 1: partial Gram  P[chunk][b] = x[b,chunk]^T @ x[b,chunk]
// One wave owns a 64x64 block (4x4 WMMA tiles): 16 dword loads per 16 WMMAs.
// Grid = B*SCH blocks of 128 threads (4 waves = 2x2 grid of 64x64 blocks).
// ---------------------------------------------------------------------------
__global__ __launch_bounds__(128) void gram_kernel(const float* __restrict__ x,
                                                   float* __restrict__ P) {
  const int lane = threadIdx.x & 31;
  const int w    = threadIdx.x >> 5;   // 0..3
  const int half = lane >> 4;          // K-group within lane
  const int m    = lane & 15;

  const int b     = blockIdx.x >> 3;   // SCH = 8 blocks per batch
  const int chunk = blockIdx.x & 7;
  const int di0   = (w >> 1) * 64;
  const int dj0   = (w & 1) * 64;

  const float* xb = x + (size_t)b * SS * DD;

  v8f acc[4][4];
#pragma unroll
  for (int i = 0; i < 4; ++i)
#pragma unroll
    for (int j = 0; j < 4; ++j) acc[i][j] = v8f{};

  const int s0 = chunk * SCHLEN;
  for (int s = s0; s < s0 + SCHLEN; s += 4) {
    const float* row0 = xb + (size_t)(s + 2 * half) * DD;
    const float* row1 = row0 + DD;
    v2f a[4], bm[4];
#pragma unroll
    for (int i = 0; i < 4; ++i) {  // A[m][k] = x[s+k][di0+16i+m]  (x^T)
      a[i].x = row0[di0 + 16 * i + m];
      a[i].y = row1[di0 + 16 * i + m];
    }
#pragma unroll
    for (int j = 0; j < 4; ++j) {  // B[k][n] = x[s+k][dj0+16j+n]
      bm[j].x = row0[dj0 + 16 * j + m];
      bm[j].y = row1[dj0 + 16 * j + m];
    }
#pragma unroll
    for (int i = 0; i < 4; ++i)
#pragma unroll
      for (int j = 0; j < 4; ++j) acc[i][j] = wmma4(a[i], bm[j], acc[i][j]);
  }

  float* Pb = P + (size_t)chunk * NBD + (size_t)b * NDD;
#pragma unroll
  for (int i = 0; i < 4; ++i)
#pragma unroll
    for (int j = 0; j < 4; ++j)
#pragma unroll
      for (int r = 0; r < 8; ++r)
        Pb[(di0 + 16 * i + r + 8 * half) * DD + dj0 + 16 * j + m] = acc[i][j][r];
}

// ---------------------------------------------------------------------------
// Phase 1b: fixed-order reduction of SCH partials; emit G in K-paired layout
//   Gp[b][(k>>1)*256 + 2n + (k&1)] = G[b][k][n]
// so the WMMA B-operand {G[k][n], G[k+1][n]} is one 8-byte load.
// ---------------------------------------------------------------------------
__global__ __launch_bounds__(256) void greduce_kernel(const float* __restrict__ P,
                                                      float* __restrict__ Gp) {
  const int i = blockIdx.x * 256 + threadIdx.x;   // over B*D*D elements
  const int b = i >> 14;
  const int r = i & (NDD - 1);
  const int k = r >> 7;
  const int n = r & (DD - 1);
  float s = 0.f;
#pragma unroll
  for (int c = 0; c < SCH; ++c) s += P[(size_t)c * NBD + i];
  Gp[(size_t)b * NDD + (k >> 1) * 256 + 2 * n + (k & 1)] = s;
}

// ---------------------------------------------------------------------------
// Phase 2a: T[b,h] = W_kq[h] @ G[b].   Grid = B*H blocks, wave w = row-tile.
// A rows contiguous (v2f); B from paired G (one b64 per lane per K-step).
// ---------------------------------------------------------------------------
__global__ __launch_bounds__(256) void mix1_kernel(const float* __restrict__ Gp,
                                                   const float* __restrict__ Wkq,
                                                   float* __restrict__ Tall) {
  const int lane = threadIdx.x & 31;
  const int w    = threadIdx.x >> 5;   // row-tile ti
  const int half = lane >> 4;
  const int m    = lane & 15;
  const int b    = blockIdx.x >> 3;
  const int h    = blockIdx.x & 7;

  const float* Gb = Gp + (size_t)b * NDD;
  const float* Wk = Wkq + (size_t)h * NDD;
  float* Tb = Tall + ((size_t)(b * HH + h)) * NDD;

  for (int tj = 0; tj < 8; ++tj) {
    v8f c = {};
    for (int kk = 0; kk < DD; kk += 4) {
      v2f a  = *(const v2f*)(Wk + (w * 16 + m) * DD + kk + 2 * half);
      v2f bm = *(const v2f*)(Gb + (kk / 2 + half) * 256 + 2 * (tj * 16 + m));
      c = wmma4(a, bm, c);
    }
#pragma unroll
    for (int r = 0; r < 8; ++r)
      Tb[(w * 16 + r + 8 * half) * DD + tj * 16 + m] = c[r];
  }
}

// ---------------------------------------------------------------------------
// Phase 2b: Mp[hg][b] = sum_{h in group hg} T[b,h] @ W_v[h]^T   (hg = 0,1)
// Both operands are contiguous v2f loads. Output in K-paired layout.
// ---------------------------------------------------------------------------
__global__ __launch_bounds__(256) void mix2_kernel(const float* __restrict__ Tall,
                                                   const float* __restrict__ Wv,
                                                   float* __restrict__ Mp) {
  const int lane = threadIdx.x & 31;
  const int w    = threadIdx.x >> 5;   // row-tile ti
  const int half = lane >> 4;
  const int m    = lane & 15;
  const int b    = blockIdx.x >> 1;
  const int hg   = blockIdx.x & 1;

  v8f acc[8];
#pragma unroll
  for (int t = 0; t < 8; ++t) acc[t] = v8f{};

  for (int hh = 0; hh < 4; ++hh) {
    const int h = hg * 4 + hh;
    const float* Tb  = Tall + ((size_t)(b * HH + h)) * NDD;
    const float* Wvh = Wv + (size_t)h * NDD;
    for (int tj = 0; tj < 8; ++tj) {
      for (int kk = 0; kk < DD; kk += 4) {
        v2f a  = *(const v2f*)(Tb + (w * 16 + m) * DD + kk + 2 * half);
        v2f bm = *(const v2f*)(Wvh + (tj * 16 + m) * DD + kk + 2 * half);
        acc[tj] = wmma4(a, bm, acc[tj]);
      }
    }
  }

  float* Mg = Mp + (size_t)hg * NBD + (size_t)b * NDD;
#pragma unroll
  for (int tj = 0; tj < 8; ++tj)
#pragma unroll
    for (int r = 0; r < 8; ++r) {
      const int row = w * 16 + r + 8 * half;
      Mg[(row >> 1) * 256 + 2 * (tj * 16 + m) + (row & 1)] = acc[tj][r];
    }
}

// ---------------------------------------------------------------------------
// Phase 3: out[b] = x[b] @ M[b].  The two head-group partials of M are summed
// (fixed order) while staging into LDS in paired layout; B-operand is then a
// single ds b64 load per lane per K-step. 8 waves x 4 row-tiles per block.
// ---------------------------------------------------------------------------
__global__ __launch_bounds__(256) void out_kernel(const float* __restrict__ x,
                                                  const float* __restrict__ Mp,
                                                  float* __restrict__ out) {
  __shared__ float Ms[NDD];            // 64 KB, K-paired layout
  const int lane   = threadIdx.x & 31;
  const int w      = threadIdx.x >> 5;
  const int half   = lane >> 4;
  const int m      = lane & 15;
  const int b      = blockIdx.x >> 3;
  const int schunk = blockIdx.x & 7;   // 512 S-rows per block

  const v4f* m0 = (const v4f*)(Mp + (size_t)b * NDD);
  const v4f* m1 = (const v4f*)(Mp + (size_t)NBD + (size_t)b * NDD);
  v4f* Ms4 = (v4f*)Ms;
  for (int i = threadIdx.x; i < NDD / 4; i += 256) Ms4[i] = m0[i] + m1[i];
  __syncthreads();

  const float* xb = x + (size_t)b * SS * DD;
  float* ob = out + (size_t)b * SS * DD;

  for (int rep = 0; rep < 4; ++rep) {
    const int si = schunk * 512 + (w * 4 + rep) * 16;
    v8f acc[8];
#pragma unroll
    for (int t = 0; t < 8; ++t) acc[t] = v8f{};

    for (int kk = 0; kk < DD; kk += 4) {
      v2f a = *(const v2f*)(xb + (size_t)(si + m) * DD + kk + 2 * half);
#pragma unroll
      for (int tj = 0; tj < 8; ++tj) {
        v2f bm = *(const v2f*)&Ms[(kk / 2 + half) * 256 + 2 * (tj * 16 + m)];
        acc[tj] = wmma4(a, bm, acc[tj]);
      }
    }
#pragma unroll
    for (int tj = 0; tj < 8; ++tj)
#pragma unroll
      for (int r = 0; r < 8; ++r)
        ob[(size_t)(si + r + 8 * half) * DD + tj * 16 + m] = acc[tj][r];
  }
}

extern "C" void kernel_launch(void* const* d_in, const int* in_sizes, int n_in,
                              void* d_out, int out_size, void* d_ws, size_t ws_size,
                              hipStream_t stream) {
  const float* x   = (const float*)d_in[0];
  const float* Wkq = (const float*)d_in[1];
  const float* Wv  = (const float*)d_in[2];
  float* out = (float*)d_out;

  float* P    = (float*)d_ws;                        // SCH * B*D*D   (16 MB)
  float* Gp   = P + (size_t)SCH * NBD;               // B*D*D (paired, 2 MB)
  float* Tall = Gp + (size_t)NBD;                    // B*H*D*D       (16 MB)
  float* Mp   = Tall + (size_t)BB * HH * NDD;        // 2 * B*D*D     (4 MB)

  gram_kernel<<<BB * SCH, 128, 0, stream>>>(x, P);
  greduce_kernel<<<NBD / 256, 256, 0, stream>>>(P, Gp);
  mix1_kernel<<<BB * HH, 256, 0, stream>>>(Gp, Wkq, Tall);
  mix2_kernel<<<BB * 2, 256, 0, stream>>>(Tall, Wv, Mp);
  out_kernel<<<BB * 8, 256, 0, stream>>>(x, Mp, out);
}